// RBF_14139032339041
// MI455X (gfx1250) — compile-verified
//
#include <hip/hip_runtime.h>

// Pairwise squared Euclidean distance on MI455X (gfx1250, wave32).
// dist[n,c] = ||x_n||^2 + ||w_c||^2 - 2 * (x @ w^T)[n,c]
// Cross term via V_WMMA_F32_16X16X4_F32 (full fp32 precision).
// Norms fused into the K-loop (computed from the same fragment loads).
// Output (512 MB) streamed with non-temporal stores to preserve L2 residency
// of x (64 MB) and prototypes (0.5 MB) inside the 192 MB L2.

typedef float v2f __attribute__((ext_vector_type(2)));
typedef float v8f __attribute__((ext_vector_type(8)));

constexpr int Nn = 131072;   // rows of x
constexpr int Cc = 1024;     // prototypes
constexpr int Dd = 128;      // feature dim
constexpr int TM = 64;       // per-wave tile rows
constexpr int TN = 64;       // per-wave tile cols
constexpr int WAVES_PER_BLOCK = 8;   // 256 threads

__global__ __launch_bounds__(256)
void rbf_dist_wmma(const float* __restrict__ x,
                   const float* __restrict__ w,
                   float* __restrict__ out) {
    const int tid  = threadIdx.x;
    const int lane = tid & 31;
    const int wib  = tid >> 5;          // wave index in block
    const int sub  = lane >> 4;         // 0: lanes 0-15, 1: lanes 16-31
    const int r    = lane & 15;

    // Tile assignment: one wave owns a 64x64 output tile.
    const int tilesN = Cc / TN;                          // 16
    const int tile   = blockIdx.x * WAVES_PER_BLOCK + wib;
    const int tn     = tile % tilesN;
    const int tm     = tile / tilesN;
    const int row0   = tm * TM;
    const int col0   = tn * TN;

    // WMMA f32 16x16x4 operand layout (ISA 7.12.2, 32-bit A 16x4):
    //   lane l, vgpr j -> A[M = l&15][K = (l>>4)*2 + j]
    //   lane l, vgpr j -> B[K = (l>>4)*2 + j][N = l&15]
    // Both A (x tile) and B (w^T tile) therefore load as one float2 per
    // 16-wide tile per k-step from row (base + (l&15)), cols (k + sub*2 .. +1).
    const float* ax = x + (size_t)(row0 + r) * Dd + sub * 2;
    const float* bw = w + (size_t)(col0 + r) * Dd + sub * 2;

    v8f acc[4][4];
    const v8f vzero = {0.f, 0.f, 0.f, 0.f, 0.f, 0.f, 0.f, 0.f};
#pragma unroll
    for (int mt = 0; mt < 4; ++mt)
#pragma unroll
        for (int nt = 0; nt < 4; ++nt)
            acc[mt][nt] = vzero;

    // Per-lane partial norms: each lane sees half of K for its row; the
    // complementary half lives in lane l^16.
    float xacc[4] = {0.f, 0.f, 0.f, 0.f};
    float wacc[4] = {0.f, 0.f, 0.f, 0.f};

#pragma unroll 4
    for (int kt = 0; kt < Dd / 4; ++kt) {
        v2f a[4], b[4];
#pragma unroll
        for (int mt = 0; mt < 4; ++mt) {
            a[mt] = *(const v2f*)(ax + (size_t)mt * 16 * Dd + kt * 4);
            xacc[mt] += a[mt].x * a[mt].x + a[mt].y * a[mt].y;
        }
#pragma unroll
        for (int nt = 0; nt < 4; ++nt) {
            b[nt] = *(const v2f*)(bw + (size_t)nt * 16 * Dd + kt * 4);
            wacc[nt] += b[nt].x * b[nt].x + b[nt].y * b[nt].y;
        }
#pragma unroll
        for (int mt = 0; mt < 4; ++mt)
#pragma unroll
            for (int nt = 0; nt < 4; ++nt)
                acc[mt][nt] = __builtin_amdgcn_wmma_f32_16x16x4_f32(
                    /*neg_a=*/false, a[mt],
                    /*neg_b=*/false, b[nt],
                    /*c_mod=*/(short)0, acc[mt][nt],
                    /*reuse_a=*/false, /*reuse_b=*/false);
    }

    // Complete the norms: lane l + lane l^16 hold complementary K halves.
    // After reduction, lane l holds ||w_{col0+nt*16+(l&15)}||^2 — exactly the
    // column this lane writes in the C/D layout.
    float wsr[4];
#pragma unroll
    for (int nt = 0; nt < 4; ++nt)
        wsr[nt] = wacc[nt] + __shfl_xor(wacc[nt], 16, 32);

    // Epilogue: C/D layout (ISA 7.12.2): vgpr v, lane l -> M = v + sub*8,
    // N = l&15. Row norm for M is held by lane M (and M+16): one bpermute.
#pragma unroll
    for (int mt = 0; mt < 4; ++mt) {
        float xsr = xacc[mt] + __shfl_xor(xacc[mt], 16, 32);
#pragma unroll
        for (int v = 0; v < 8; ++v) {
            const int mloc = v + sub * 8;                 // 0..15
            const float xv = __shfl(xsr, mloc, 32);       // row norm broadcast
            const size_t row = (size_t)(row0 + mt * 16 + mloc);
            float* orow = out + row * Cc + col0 + r;
#pragma unroll
            for (int nt = 0; nt < 4; ++nt) {
                const float d = xv + wsr[nt] - 2.0f * acc[mt][nt][v];
                __builtin_nontemporal_store(d, orow + nt * 16);
            }
        }
    }
}

extern "C" void kernel_launch(void* const* d_in, const int* in_sizes, int n_in,
                              void* d_out, int out_size, void* d_ws, size_t ws_size,
                              hipStream_t stream) {
    const float* x = (const float*)d_in[0];   // [N, D] fp32
    const float* w = (const float*)d_in[1];   // [C, D] fp32
    float* out = (float*)d_out;               // [N, C] fp32

    const int tiles = (Nn / TM) * (Cc / TN);  // 2048 * 16 = 32768 waves
    dim3 block(32 * WAVES_PER_BLOCK);         // 256 threads = 8 waves
    dim3 grid(tiles / WAVES_PER_BLOCK);       // 4096 blocks
    rbf_dist_wmma<<<grid, block, 0, stream>>>(x, w, out);
}